// Attention_884763263569
// MI455X (gfx1250) — compile-verified
//
#include <hip/hip_runtime.h>
#include <hip/hip_bf16.h>
#include <math.h>

// CDNA5 (gfx1250) fused attention-pool kernel:
//   k_t = x_t @ Wk^T, v_t = x_t @ Wv^T, q = concat(x) @ Wq^T,
//   a = softmax_t(q . k_t), c = sum_t a_t v_t, out = c @ Wfc^T + bfc
// B=65536, T=3, D=512, P=32, C=10. One wave32 per block, 16 samples/wave.
// x tiles are streamed into LDS by the Tensor Data Mover (double-buffered,
// TENSORcnt-tracked) and consumed by V_WMMA_F32_16X16X4_F32.

typedef __attribute__((ext_vector_type(2))) float v2f;
typedef __attribute__((ext_vector_type(8))) float v8f;
typedef __attribute__((ext_vector_type(4))) unsigned int u32x4;
typedef __attribute__((ext_vector_type(8))) int i32x8;
typedef __attribute__((ext_vector_type(4))) int i32x4;

#define T_DIM 3
#define D_DIM 512
#define P_DIM 32
#define C_DIM 10
// LDS row layout: 512 data dwords + 2-dword pad after each 256 dwords
// (TDM: pad_interval=256 dw, pad_amount=2 dw) -> stride 516 dwords.
// ds_load_b64 fragment reads then hit all 64 banks conflict-free.
#define ROW_STRIDE 516
#define TILE_DWORDS (16 * ROW_STRIDE)

#if __has_builtin(__builtin_amdgcn_tensor_load_to_lds) && __has_builtin(__builtin_amdgcn_s_wait_tensorcnt)
#define USE_TDM 1
#else
#define USE_TDM 0
#endif

__device__ __forceinline__ unsigned lds_byte_offset(const void* p) {
    // generic -> LDS address space; AS(3) pointers are 32-bit LDS byte offsets
    return (unsigned)(__SIZE_TYPE__)(__attribute__((address_space(3))) const void*)p;
}

#if USE_TDM
// DMA one 16x512 f32 tile (row stride 1536 floats in memory) into LDS with
// 2-dword padding every 256 dwords. Descriptor per CDNA5 ISA.
// This toolchain exposes the 6-arg builtin:
//   (uint32x4 g0, int32x8 g1, int32x4 g2, int32x4 g3, int32x8 extra, i32 cpol)
__device__ __forceinline__ void tdm_load_tile(const float* gsrc, unsigned lds_off) {
    unsigned long long ga = (unsigned long long)gsrc;
    u32x4 g0 = {
        1u,                                                // count=1 (valid user D#)
        lds_off,                                           // lds_addr (bytes)
        (unsigned)(ga & 0xFFFFFFFFu),                      // global_addr[31:0]
        (unsigned)((ga >> 32) & 0x01FFFFFFu) | (2u << 30)  // global_addr[56:32] | type=2
    };
    i32x8 g1 = {
        (2 << 16) | (1 << 20) | (7 << 22) | (1 << 25), // data_size=4B, pad_en, interval=256dw, amount=2dw
        (int)(512u << 16),                             // tensor_dim0 = 512 (lo16 in [31:16])
        (int)(16u << 16),                              // tensor_dim0 hi=0 | tensor_dim1=16 lo16
        (int)(512u << 16),                             // tensor_dim1 hi=0 | tile_dim0=512
        16,                                            // tile_dim1=16, tile_dim2=0
        1536,                                          // tensor_dim0_stride[31:0] = T*D
        0, 0                                           // stride hi / dim1_stride (unused, 2D)
    };
    i32x4 gz4 = {0, 0, 0, 0};                          // groups 2/3 unused (<=2D tensor)
    i32x8 gz8 = {0, 0, 0, 0, 0, 0, 0, 0};              // trailing descriptor group (zero-filled)
    __builtin_amdgcn_tensor_load_to_lds(g0, g1, gz4, gz4, gz8, 0);
}
#endif

// One K-step: A fragment from LDS (constant offsets within the half-loop) and
// six B fragments feeding six V_WMMA_F32_16X16X4_F32.
__device__ __forceinline__ void gemm_step(
    const float* __restrict__ tilep,   // tile base, pad already folded in (constant)
    const float* __restrict__ Wk, const float* __restrict__ Wv,
    const float* __restrict__ Wq,
    int m, int koff, int kk, int t,
    v8f& k0, v8f& k1, v8f& v0, v8f& v1, v8f& q0, v8f& q1)
{
    // A fragment (16x4 f32): lanes 0-15 -> K=kk,kk+1 ; lanes 16-31 -> K=kk+2,kk+3
    v2f a = *(const v2f*)(tilep + m * ROW_STRIDE + kk + koff);

    // B fragments (4x16) of Wk^T / Wv^T / Wq^T; lane -> N=m (tile0) / m+16 (tile1)
    const float* wkp = Wk + (size_t)m * D_DIM + kk + koff;
    v2f bk0 = *(const v2f*)(wkp);
    v2f bk1 = *(const v2f*)(wkp + 16 * D_DIM);
    const float* wvp = Wv + (size_t)m * D_DIM + kk + koff;
    v2f bv0 = *(const v2f*)(wvp);
    v2f bv1 = *(const v2f*)(wvp + 16 * D_DIM);
    const float* wqp = Wq + (size_t)m * (T_DIM * D_DIM) + t * D_DIM + kk + koff;
    v2f bq0 = *(const v2f*)(wqp);
    v2f bq1 = *(const v2f*)(wqp + 16 * (T_DIM * D_DIM));

    k0 = __builtin_amdgcn_wmma_f32_16x16x4_f32(false, a, false, bk0, (short)0, k0, false, false);
    k1 = __builtin_amdgcn_wmma_f32_16x16x4_f32(false, a, false, bk1, (short)0, k1, false, false);
    v0 = __builtin_amdgcn_wmma_f32_16x16x4_f32(false, a, false, bv0, (short)0, v0, false, false);
    v1 = __builtin_amdgcn_wmma_f32_16x16x4_f32(false, a, false, bv1, (short)0, v1, false, false);
    q0 = __builtin_amdgcn_wmma_f32_16x16x4_f32(false, a, false, bq0, (short)0, q0, false, false);
    q1 = __builtin_amdgcn_wmma_f32_16x16x4_f32(false, a, false, bq1, (short)0, q1, false, false);
}

__global__ __launch_bounds__(32) void attn_fused_kernel(
    const float* __restrict__ x,    // [B, T, D]
    const float* __restrict__ Wk,   // [P, D]
    const float* __restrict__ Wv,   // [P, D]
    const float* __restrict__ Wq,   // [P, T*D]
    const float* __restrict__ Wfc,  // [C, P]
    const float* __restrict__ bfc,  // [C]
    float* __restrict__ out)        // [B, C]
{
    __shared__ float xs[2 * TILE_DWORDS];  // double-buffered x tiles
    __shared__ float cs[16 * P_DIM];       // context vectors for FC stage

    const int lane = threadIdx.x;          // 0..31 (wave32)
    const int base = blockIdx.x * 16;      // 16 samples per block
    const int m    = lane & 15;
    const int half = lane >> 4;
    const int koff = half * 2;

    // Accumulators in 16x16 f32 C/D layout (VGPR r, lane L: M=r+8*(L/16), N=L%16 [+16])
    v8f qa[2]        = {};
    v8f ka[T_DIM][2] = {};
    v8f va[T_DIM][2] = {};

#if USE_TDM
    // Prime the pipeline: DMA view 0 into buffer 0.
    tdm_load_tile(x + (size_t)base * (T_DIM * D_DIM), lds_byte_offset(xs));
#endif

#pragma unroll
    for (int t = 0; t < T_DIM; ++t) {
        const float* tile = xs + (t & 1) * TILE_DWORDS;

#if USE_TDM
        if (t + 1 < T_DIM) {
            // WAR guard: buffer (t+1)&1 was read in the previous iteration; all of this
            // wave's ds reads must have completed before the DMA overwrites it.
            asm volatile("s_wait_dscnt 0x0" ::: "memory");
            tdm_load_tile(x + ((size_t)base * T_DIM + (t + 1)) * D_DIM,
                          lds_byte_offset(xs + ((t + 1) & 1) * TILE_DWORDS));
            // Wait until view t's DMA is complete (t+1's still in flight).
            __builtin_amdgcn_s_wait_tensorcnt(1);
        } else {
            __builtin_amdgcn_s_wait_tensorcnt(0);
        }
        __syncthreads();
#else
        // Fallback: manual coalesced staging with the same padded layout.
        float* wtile = (float*)tile;
#pragma unroll 4
        for (int j = 0; j < 64; ++j) {
            int idx = j * 32 + lane;            // 2048 float4 chunks
            int row = idx >> 7;
            int c   = (idx & 127) * 4;
            const float4* src =
                (const float4*)(x + ((size_t)(base + row) * T_DIM + t) * D_DIM + c);
            *(float4*)(wtile + row * ROW_STRIDE + c + ((c >> 8) << 1)) = *src;
        }
        if (t + 1 < T_DIM)
            __builtin_prefetch(x + ((size_t)(base + m) * T_DIM + (t + 1)) * D_DIM + half * 256, 0, 1);
        __syncthreads();
#endif

        // ---- GEMM phase, split at the pad boundary so the 2-dword pad folds
        //      into the base pointer as a constant (restores ds_load_2addr_b64
        //      with immediate offsets; no per-step VALU address math). ----
#pragma unroll 4
        for (int kk = 0; kk < 256; kk += 4)       // cols kk+koff <= 254: no pad
            gemm_step(tile, Wk, Wv, Wq, m, koff, kk, t,
                      ka[t][0], ka[t][1], va[t][0], va[t][1], qa[0], qa[1]);
#pragma unroll 4
        for (int kk = 256; kk < 512; kk += 4)     // cols >= 256: +2-dword pad
            gemm_step(tile + 2, Wk, Wv, Wq, m, koff, kk, t,
                      ka[t][0], ka[t][1], va[t][0], va[t][1], qa[0], qa[1]);

#if !USE_TDM
        __syncthreads();  // tile about to be overwritten next iteration
#endif
    }

    // ---- Logits s[m][t] = sum_n q[m][n]*k_t[m][n]: elementwise + 16-lane reduce ----
    float s[8][T_DIM];
#pragma unroll
    for (int t = 0; t < T_DIM; ++t) {
        v8f prod = qa[0] * ka[t][0] + qa[1] * ka[t][1];
#pragma unroll
        for (int r = 0; r < 8; ++r) {
            float p = prod[r];
            p += __shfl_xor(p, 1, 32);
            p += __shfl_xor(p, 2, 32);
            p += __shfl_xor(p, 4, 32);
            p += __shfl_xor(p, 8, 32);   // stays within the 16-lane half (M = r + 8*half)
            s[r][t] = p;
        }
    }

    // ---- 3-way softmax + weighted sum of v (registers only) ----
    v8f c0 = {}, c1 = {};
#pragma unroll
    for (int r = 0; r < 8; ++r) {
        float mx = fmaxf(s[r][0], fmaxf(s[r][1], s[r][2]));
        float e0 = __expf(s[r][0] - mx);
        float e1 = __expf(s[r][1] - mx);
        float e2 = __expf(s[r][2] - mx);
        float inv = 1.0f / (e0 + e1 + e2);
        float a0 = e0 * inv, a1 = e1 * inv, a2 = e2 * inv;
        c0[r] = a0 * va[0][0][r] + a1 * va[1][0][r] + a2 * va[2][0][r];
        c1[r] = a0 * va[0][1][r] + a1 * va[1][1][r] + a2 * va[2][1][r];
    }

    // ---- Spill c (16 x 32) to LDS, then tiny FC: out = c @ Wfc^T + bfc ----
#pragma unroll
    for (int r = 0; r < 8; ++r) {
        cs[(r + 8 * half) * P_DIM + m]      = c0[r];
        cs[(r + 8 * half) * P_DIM + m + 16] = c1[r];
    }
    __syncthreads();

    for (int o = lane; o < 16 * C_DIM; o += 32) {
        int mm  = o / C_DIM;
        int cls = o % C_DIM;
        float acc = bfc[cls];
#pragma unroll
        for (int n = 0; n < P_DIM; ++n)
            acc += cs[mm * P_DIM + n] * Wfc[cls * P_DIM + n];
        out[(size_t)(base + mm) * C_DIM + cls] = acc;
    }
}

extern "C" void kernel_launch(void* const* d_in, const int* in_sizes, int n_in,
                              void* d_out, int out_size, void* d_ws, size_t ws_size,
                              hipStream_t stream) {
    const float* x   = (const float*)d_in[0];
    const float* Wk  = (const float*)d_in[1];
    const float* Wv  = (const float*)d_in[2];
    const float* Wq  = (const float*)d_in[3];
    const float* Wfc = (const float*)d_in[4];
    const float* bfc = (const float*)d_in[5];
    float* out = (float*)d_out;

    int Bn = in_sizes[0] / (T_DIM * D_DIM);   // 65536
    dim3 grid(Bn / 16);
    attn_fused_kernel<<<grid, 32, 0, stream>>>(x, Wk, Wv, Wq, Wfc, bfc, out);
}